// DiffusionModel1d_77429670412757
// MI455X (gfx1250) — compile-verified
//
#include <hip/hip_runtime.h>
#include <stdint.h>

// Batched Thomas tridiagonal solve, one wave32 per row.
// N=4096, BATCH=8192. Memory-bound: ideal traffic 256MB @ 23.3TB/s ~= 11us.
// c/u intermediates kept entirely in LDS (~152KB/block, 2 blocks per 320KB WGP)
// to avoid a 512MB HBM round-trip of scratch. x tiles staged into LDS with
// gfx1250 async loads (global_load_async_to_lds_b128 + s_wait_asynccnt),
// prefetch depth 2 (4 buffers). The serial recurrences are parallelized across
// the wave as Mobius (2x2) / affine scans with 4 elements folded per lane.

#define NN      4096
#define BATCH   8192
#define CH      128            // elements per chunk (one wave iteration)
#define NCH     (NN / CH)      // 32 chunks per row
#define WAVES   4              // rows per block
#define THREADS (WAVES * 32)
#define NBUF    4              // staging buffers per wave (prefetch depth 2)
#define LDS_FLOATS (NN + WAVES * 2 * NN + WAVES * NBUF * CH)
#define LDS_BYTES  (LDS_FLOATS * 4)

__device__ __forceinline__ float rcpf(float x) { return __builtin_amdgcn_rcpf(x); }

// Flat address of an LDS pointer carries the LDS byte offset in its low 32 bits
// (ISA 10.2).
__device__ __forceinline__ uint32_t lds_off(const void* p) {
  return (uint32_t)(uintptr_t)p;
}

// One instruction stages 32 lanes x 16B = 512B of x into LDS (ASYNCcnt path).
__device__ __forceinline__ void async_load_b128(uint32_t lds_byte, const void* g) {
  asm volatile("global_load_async_to_lds_b128 %0, %1, off"
               :: "v"(lds_byte), "v"((uint64_t)(uintptr_t)g)
               : "memory");
}
__device__ __forceinline__ void wait_async_0() {
  asm volatile("s_wait_asynccnt 0" ::: "memory");
}
__device__ __forceinline__ void wait_async_1() {
  asm volatile("s_wait_asynccnt 1" ::: "memory");
}
__device__ __forceinline__ void wait_async_2() {
  asm volatile("s_wait_asynccnt 2" ::: "memory");
}

// Fold elementary Mobius matrix E=[[0,-K1],[K,K+K1]] into accumulator: P = E*P.
#define MOB_FOLD(Kk, Kk1)                                                     \
  do {                                                                        \
    float n00 = -(Kk1) * p10, n01 = -(Kk1) * p11;                             \
    float n10 = (Kk) * p00 + ((Kk) + (Kk1)) * p10;                            \
    float n11 = (Kk) * p01 + ((Kk) + (Kk1)) * p11;                            \
    p00 = n00; p01 = n01; p10 = n10; p11 = n11;                               \
  } while (0)

__global__ __launch_bounds__(THREADS)
void thomas_rows_kernel(const float* __restrict__ x,
                        const float* __restrict__ f_rhs,
                        float* __restrict__ out) {
  extern __shared__ __align__(16) float smem[];
  const int tid  = threadIdx.x;
  const int wave = tid >> 5;
  const int lane = tid & 31;
  const int row  = blockIdx.x * WAVES + wave;

  float* fbuf = smem;                                    // [NN] shared f
  float* cbuf = smem + NN + wave * (2 * NN);             // [NN] per wave
  float* ubuf = cbuf + NN;                               // [NN] per wave
  float* stag = smem + NN + WAVES * (2 * NN) + wave * (NBUF * CH);

  const float* xrow = x + (size_t)row * NN;
  float* outrow = out + (size_t)row * NN;

  // Kick off the async pipeline immediately: prefetch chunks 0 and 1.
  async_load_b128(lds_off(stag + lane * 4), xrow + lane * 4);
  async_load_b128(lds_off(stag + CH + lane * 4), xrow + CH + lane * 4);

  // Cooperative float4 preload of f_rhs into LDS, shared by all 4 rows.
  for (int i = tid * 4; i < NN; i += THREADS * 4)
    *(float4*)(fbuf + i) = *(const float4*)(f_rhs + i);
  __syncthreads();

  // ---- forward sweep: uniform recurrence with K[NN]:=0, carry (c,u)=(1,0) ----
  // denom_j = K_j + K_{j+1} + K_j*c_{j-1}; c_j = -K_{j+1}/denom;
  // u_j = (f_j + K_j*u_{j-1})/denom
  float c_carry = 1.0f, u_carry = 0.0f;

  for (int t = 0; t < NCH; ++t) {
    const int base = t * CH;
    float* buf = stag + (t & (NBUF - 1)) * CH;
    if (t + 2 < NCH) {
      float* nbuf = stag + ((t + 2) & (NBUF - 1)) * CH;
      async_load_b128(lds_off(nbuf + lane * 4), xrow + (t + 2) * CH + lane * 4);
      wait_async_2();                 // chunk t staged; t+1, t+2 in flight
    } else if (t + 1 < NCH) {
      wait_async_1();
    } else {
      wait_async_0();
    }

    // K for this chunk's boundary element (K[base+CH]); 0 past the end.
    float kedge = 0.0f;
    const int eidx = base + CH;
    if (eidx < NN) kedge = __expf(xrow[eidx]);

    const int s = lane * 4;
    float4 xv = *(const float4*)(buf + s);
    float K0 = __expf(xv.x), K1 = __expf(xv.y), K2 = __expf(xv.z), K3 = __expf(xv.w);
    float K4 = __shfl_down(K0, 1, 32);
    if (lane == 31) K4 = kedge;

    // --- lane-local Mobius composite for c over 4 elements: E3*E2*E1*E0 ---
    float p00 = 0.0f, p01 = -K1, p10 = K0, p11 = K0 + K1;
    MOB_FOLD(K1, K2);
    MOB_FOLD(K2, K3);
    MOB_FOLD(K3, K4);
    {
      float sc = rcpf(fabsf(p10) + fabsf(p11) + 1e-30f);
      p00 *= sc; p01 *= sc; p10 *= sc; p11 *= sc;
    }

    // --- cross-lane inclusive scan of Mobius composites (P = P_cur * P_fetched) ---
    #pragma unroll
    for (int d = 1; d < 32; d <<= 1) {
      float q00 = __shfl_up(p00, d, 32), q01 = __shfl_up(p01, d, 32);
      float q10 = __shfl_up(p10, d, 32), q11 = __shfl_up(p11, d, 32);
      if (lane >= d) {
        float n00 = p00 * q00 + p01 * q10, n01 = p00 * q01 + p01 * q11;
        float n10 = p10 * q00 + p11 * q10, n11 = p10 * q01 + p11 * q11;
        float sc = rcpf(fabsf(n10) + fabsf(n11) + 1e-30f);
        p00 = n00 * sc; p01 = n01 * sc; p10 = n10 * sc; p11 = n11 * sc;
      }
    }
    // exclusive prefix -> incoming c for this lane
    float e00 = __shfl_up(p00, 1, 32), e01 = __shfl_up(p01, 1, 32);
    float e10 = __shfl_up(p10, 1, 32), e11 = __shfl_up(p11, 1, 32);
    float c_in = (lane == 0) ? c_carry
                             : (e00 * c_carry + e01) * rcpf(e10 * c_carry + e11);
    float i00 = __shfl(p00, 31, 32), i01 = __shfl(p01, 31, 32);
    float i10 = __shfl(p10, 31, 32), i11 = __shfl(p11, 31, 32);
    float c_carry_next = (i00 * c_carry + i01) * rcpf(i10 * c_carry + i11);

    // --- serial c + build lane affine composite for u ---
    float4 fvv = *(const float4*)(fbuf + base + s);
    float Ks[5] = {K0, K1, K2, K3, K4};
    float fs[4] = {fvv.x, fvv.y, fvv.z, fvv.w};
    float cl[4], rl[4];
    float cp = c_in, Al = 1.0f, Bl = 0.0f;
    #pragma unroll
    for (int k = 0; k < 4; ++k) {
      float denom = Ks[k] + Ks[k + 1] + Ks[k] * cp;
      float r = rcpf(denom);
      rl[k] = r;
      float c = -Ks[k + 1] * r;
      cl[k] = c; cp = c;
      float Ak = Ks[k] * r, Bk = fs[k] * r;
      Bl = Ak * Bl + Bk;      // compose T_k after accumulator
      Al = Ak * Al;
    }

    // --- cross-lane inclusive affine scan for u ---
    float A = Al, B = Bl;
    #pragma unroll
    for (int d = 1; d < 32; d <<= 1) {
      float Aq = __shfl_up(A, d, 32), Bq = __shfl_up(B, d, 32);
      if (lane >= d) { B = A * Bq + B; A = A * Aq; }
    }
    float Ae = __shfl_up(A, 1, 32), Be = __shfl_up(B, 1, 32);
    float u_in = (lane == 0) ? u_carry : (Ae * u_carry + Be);
    float A31 = __shfl(A, 31, 32), B31 = __shfl(B, 31, 32);
    float u_carry_next = A31 * u_carry + B31;

    float ul[4];
    float up = u_in;
    #pragma unroll
    for (int k = 0; k < 4; ++k) {
      float u = (fs[k] + Ks[k] * up) * rl[k];
      ul[k] = u; up = u;
    }

    *(float4*)(cbuf + base + s) = make_float4(cl[0], cl[1], cl[2], cl[3]);
    *(float4*)(ubuf + base + s) = make_float4(ul[0], ul[1], ul[2], ul[3]);

    c_carry = c_carry_next;
    u_carry = u_carry_next;
  }

  // ---- backward substitution: u_j = u_j - c_j * u_{j+1}, affine reverse scan ----
  float ucb = 0.0f; // c[N-1]==0 makes the seed irrelevant for the last element
  for (int t = NCH - 1; t >= 0; --t) {
    const int base = t * CH;
    const int s = lane * 4;
    float4 c4 = *(const float4*)(cbuf + base + s);
    float4 u4 = *(const float4*)(ubuf + base + s);

    // lane composite G_{j0} o G_{j0+1} o G_{j0+2} o G_{j0+3}, G_j(v) = -c_j*v + u_j
    float A = -c4.w, B = u4.w;
    A = -c4.z * A;  B = -c4.z * B + u4.z;
    A = -c4.y * A;  B = -c4.y * B + u4.y;
    A = -c4.x * A;  B = -c4.x * B + u4.x;

    #pragma unroll
    for (int d = 1; d < 32; d <<= 1) {
      float Aq = __shfl_down(A, d, 32), Bq = __shfl_down(B, d, 32);
      if (lane + d < 32) { B = A * Bq + B; A = A * Aq; }
    }
    float Ae = __shfl_down(A, 1, 32), Be = __shfl_down(B, 1, 32);
    float v_in = (lane == 31) ? ucb : (Ae * ucb + Be);
    float A0 = __shfl(A, 0, 32), B0 = __shfl(B, 0, 32);
    float carry_next = A0 * ucb + B0;

    float o3 = u4.w - c4.w * v_in;
    float o2 = u4.z - c4.z * o3;
    float o1 = u4.y - c4.y * o2;
    float o0 = u4.x - c4.x * o1;
    *(float4*)(outrow + base + s) = make_float4(o0, o1, o2, o3);

    ucb = carry_next;
  }
}

extern "C" void kernel_launch(void* const* d_in, const int* in_sizes, int n_in,
                              void* d_out, int out_size, void* d_ws, size_t ws_size,
                              hipStream_t stream) {
  (void)in_sizes; (void)n_in; (void)out_size; (void)d_ws; (void)ws_size;
  const float* x = (const float*)d_in[0];
  const float* f = (const float*)d_in[1];
  float* out = (float*)d_out;
  dim3 grid(BATCH / WAVES), block(THREADS);
  hipLaunchKernelGGL(thomas_rows_kernel, grid, block, LDS_BYTES, stream, x, f, out);
}